// DeepPolyReluLayer_59914793779496
// MI455X (gfx1250) — compile-verified
//
#include <hip/hip_runtime.h>
#include <hip/hip_bf16.h>

// DeepPoly ReLU bound propagation, elementwise over N neurons.
//   x_out = max(x, 0)
//   neg  = (u <= 0)          -> bounds 0
//   pos  = (l >= 0)          -> both bounds u
//   else (crossing)          -> lower_out = l, upper_out = u * (u / (u - l))
//
// Memory-bound: 24 B/elem streamed once each way (~402 MB total, ~17 us floor
// at 23.3 TB/s). Use native clang 4-float vectors for 128-bit (b128) accesses
// with non-temporal cache policy (CDNA5 TH=NT) since there is zero reuse and
// the working set exceeds the 192 MB L2.

typedef float v4f __attribute__((ext_vector_type(4)));

__device__ __forceinline__ void deeppoly_relu_elem(float x, float l, float u,
                                                   float& xo, float& lo, float& uo) {
    const bool neg = (u <= 0.0f);
    const bool pos = (l >= 0.0f);
    const bool crossing = (!neg) && (!pos);

    xo = fmaxf(x, 0.0f);

    // safe slope: denom = 1 unless crossing (denom > 0 when crossing)
    const float denom = crossing ? (u - l) : 1.0f;
    const float slope = u / denom;              // IEEE divide, matches jnp

    lo = neg ? 0.0f : (pos ? u : l);
    uo = neg ? 0.0f : (pos ? u : slope * u);
}

__global__ void __launch_bounds__(256)
DeepPolyReluLayer_59914793779496_kernel(const float* __restrict__ x,
                                        const float* __restrict__ lower,
                                        const float* __restrict__ upper,
                                        float* __restrict__ out,   // [3*n]: x_out | lower_out | upper_out
                                        int n) {
    const long long tid = (long long)blockIdx.x * blockDim.x + threadIdx.x;
    const long long n4  = (long long)n >> 2;   // number of full float4 groups

    float* __restrict__ x_out = out;
    float* __restrict__ l_out = out + n;
    float* __restrict__ u_out = out + 2LL * n;

    if (tid < n4) {
        const long long base = tid * 4;

        // 128-bit non-temporal streaming loads (global_load_b128, TH=NT)
        const v4f xv = __builtin_nontemporal_load((const v4f*)(x     + base));
        const v4f lv = __builtin_nontemporal_load((const v4f*)(lower + base));
        const v4f uv = __builtin_nontemporal_load((const v4f*)(upper + base));

        v4f xo, lo, uo;
        #pragma unroll
        for (int j = 0; j < 4; ++j) {
            float a, b, c;
            deeppoly_relu_elem(xv[j], lv[j], uv[j], a, b, c);
            xo[j] = a; lo[j] = b; uo[j] = c;
        }

        // 128-bit non-temporal streaming stores (global_store_b128, TH=NT)
        __builtin_nontemporal_store(xo, (v4f*)(x_out + base));
        __builtin_nontemporal_store(lo, (v4f*)(l_out + base));
        __builtin_nontemporal_store(uo, (v4f*)(u_out + base));
    }

    // Scalar tail (n not divisible by 4) — handled by the first few threads.
    const long long tail_start = n4 * 4;
    const long long tail_cnt   = (long long)n - tail_start;
    if (tid < tail_cnt) {
        const long long i = tail_start + tid;
        float xo, lo, uo;
        deeppoly_relu_elem(x[i], lower[i], upper[i], xo, lo, uo);
        __builtin_nontemporal_store(xo, x_out + i);
        __builtin_nontemporal_store(lo, l_out + i);
        __builtin_nontemporal_store(uo, u_out + i);
    }
}

extern "C" void kernel_launch(void* const* d_in, const int* in_sizes, int n_in,
                              void* d_out, int out_size, void* d_ws, size_t ws_size,
                              hipStream_t stream) {
    (void)n_in; (void)out_size; (void)d_ws; (void)ws_size;

    const float* x     = (const float*)d_in[0];
    const float* lower = (const float*)d_in[1];
    const float* upper = (const float*)d_in[2];
    float* out = (float*)d_out;

    const int n = in_sizes[0];                 // 16,777,216
    const long long n4 = ((long long)n + 3) >> 2;

    const int block = 256;                     // 8 wave32 waves per block
    const long long grid = (n4 + block - 1) / block;

    DeepPolyReluLayer_59914793779496_kernel<<<(unsigned)grid, block, 0, stream>>>(
        x, lower, upper, out, n);
}